// LightGCNModel_17377437680517
// MI455X (gfx1250) — compile-verified
//
#include <hip/hip_runtime.h>

#define N_USERS_C 100000
#define N_ITEMS_C 50000
#define NNODES (N_USERS_C + N_ITEMS_C)
#define DIM 64
#define NLAYERS 3

typedef float v2f __attribute__((ext_vector_type(2)));
typedef float v8f __attribute__((ext_vector_type(8)));

// ---- init: cur = embeds, acc = embeds --------------------------------------
__global__ void init_copy(const float4* __restrict__ src, float4* __restrict__ cur,
                          float4* __restrict__ acc, int n4) {
  int i = blockIdx.x * blockDim.x + threadIdx.x;
  if (i < n4) { float4 v = src[i]; cur[i] = v; acc[i] = v; }
}

// ---- zero the destination buffer -------------------------------------------
__global__ void zero_buf(float4* __restrict__ p, int n4) {
  int i = blockIdx.x * blockDim.x + threadIdx.x;
  if (i < n4) p[i] = make_float4(0.f, 0.f, 0.f, 0.f);
}

// ---- SpMM scatter: one wave32 per edge, lane l handles dims (2l,2l+1) ------
__global__ void spmm_scatter(const float* __restrict__ cur, float* __restrict__ nxt,
                             const float* __restrict__ vals, const int* __restrict__ rows,
                             const int* __restrict__ cols, int E) {
  int gid = blockIdx.x * blockDim.x + threadIdx.x;
  int e = gid >> 5;
  if (e >= E) return;
  int lane = threadIdx.x & 31;
  int row = rows[e];
  int col = cols[e];
  float v = vals[e];
  float2 m = ((const float2*)(cur + (size_t)col * DIM))[lane];   // coalesced 256B row
  float* dst = nxt + (size_t)row * DIM + lane * 2;
  __hip_atomic_fetch_add(dst + 0, v * m.x, __ATOMIC_RELAXED, __HIP_MEMORY_SCOPE_AGENT);
  __hip_atomic_fetch_add(dst + 1, v * m.y, __ATOMIC_RELAXED, __HIP_MEMORY_SCOPE_AGENT);
}

// ---- acc += nxt via exact identity-matmul on the matrix pipe ----------------
// One wave per 16x16 tile. D = sum_k A_chunk(16x4) x I_chunk(4x16) + C == nxt_tile + acc_tile.
// Products are x*1.0f or x*0.0f in f32 => bit-exact vs scalar add.
__global__ void wmma_accumulate(float* __restrict__ acc, const float* __restrict__ nxt,
                                int nTileRows) {
  int wave = (blockIdx.x * blockDim.x + threadIdx.x) >> 5;
  int lane = threadIdx.x & 31;
  int trow = wave >> 2;          // 4 column tiles across DIM=64
  int tcol = wave & 3;
  if (trow >= nTileRows) return; // wave-uniform: EXEC stays all-ones for active waves
  const int rowbase = trow * 16;
  const int colbase = tcol * 16;
  const int lhi = lane >> 4;     // 0 or 1
  const int llo = lane & 15;

  // C/D layout: VGPR j, lanes 0-15 -> M=j, lanes 16-31 -> M=j+8; N = lane%16
  v8f c;
#pragma unroll
  for (int j = 0; j < 8; ++j)
    c[j] = acc[(size_t)(rowbase + j + 8 * lhi) * DIM + colbase + llo];

#pragma unroll
  for (int ch = 0; ch < 4; ++ch) {
    // A layout (16x4 f32): lane l -> M = l%16; VGPR j -> K = 2*(l/16) + j
    v2f a, b;
#pragma unroll
    for (int j = 0; j < 2; ++j) {
      int k = 2 * lhi + j;                       // local K in 0..3
      a[j] = nxt[(size_t)(rowbase + llo) * DIM + colbase + 4 * ch + k];
      // B (4x16) identity chunk: B[k][n] = (n == 4*ch + k)
      b[j] = (llo == 4 * ch + k) ? 1.0f : 0.0f;
    }
    c = __builtin_amdgcn_wmma_f32_16x16x4_f32(
        /*neg_a=*/false, a, /*neg_b=*/false, b,
        /*c_mod=*/(short)0, c, /*reuse_a=*/false, /*reuse_b=*/false);
  }

#pragma unroll
  for (int j = 0; j < 8; ++j)
    acc[(size_t)(rowbase + j + 8 * lhi) * DIM + colbase + llo] = c[j];
}

// ---- final gather: out = 0.25 * acc[node] for users / pos / neg -------------
__global__ void gather_out(const float* __restrict__ acc, const int* __restrict__ users,
                           const int* __restrict__ pos, const int* __restrict__ neg,
                           float* __restrict__ out, int B) {
  int gid = blockIdx.x * blockDim.x + threadIdx.x;   // one float4 per thread
  int total = 3 * B * (DIM / 4);
  if (gid >= total) return;
  int r = gid / (DIM / 4);
  int q = gid % (DIM / 4);
  int which = r / B;
  int b = r - which * B;
  int node = (which == 0) ? users[b]
           : (which == 1) ? (N_USERS_C + pos[b])
                          : (N_USERS_C + neg[b]);
  float4 v = ((const float4*)(acc + (size_t)node * DIM))[q];
  const float s = 1.0f / (NLAYERS + 1);
  v.x *= s; v.y *= s; v.z *= s; v.w *= s;
  ((float4*)out)[gid] = v;
}

extern "C" void kernel_launch(void* const* d_in, const int* in_sizes, int n_in,
                              void* d_out, int out_size, void* d_ws, size_t ws_size,
                              hipStream_t stream) {
  const float* embeds = (const float*)d_in[0];
  const float* vals   = (const float*)d_in[1];
  const int*   rows   = (const int*)d_in[2];
  const int*   cols   = (const int*)d_in[3];
  const int*   users  = (const int*)d_in[4];
  const int*   pos    = (const int*)d_in[5];
  const int*   neg    = (const int*)d_in[6];
  float* out = (float*)d_out;

  const int E = in_sizes[1];
  const int B = in_sizes[4];

  const size_t nd = (size_t)NNODES * DIM;       // 9.6M floats per buffer
  float* bufA = (float*)d_ws;                   // cur (layer 0)
  float* bufB = bufA + nd;                      // nxt (layer 0)
  float* acc  = bufB + nd;                      // running sum
  const int n4  = (int)(nd / 4);
  const int blk = 256;

  init_copy<<<(n4 + blk - 1) / blk, blk, 0, stream>>>(
      (const float4*)embeds, (float4*)bufA, (float4*)acc, n4);

  const float* cur = bufA;
  float* nxt = bufB;

  const int spmmBlocks = (int)(((size_t)E * 32 + blk - 1) / blk);
  const int nTileRows  = NNODES / 16;           // 9375 (exact)
  const int wmmaWaves  = nTileRows * (DIM / 16);
  const int wmmaBlocks = (wmmaWaves * 32 + blk - 1) / blk;

  for (int layer = 0; layer < NLAYERS; ++layer) {
    zero_buf<<<(n4 + blk - 1) / blk, blk, 0, stream>>>((float4*)nxt, n4);
    spmm_scatter<<<spmmBlocks, blk, 0, stream>>>(cur, nxt, vals, rows, cols, E);
    wmma_accumulate<<<wmmaBlocks, blk, 0, stream>>>(acc, nxt, nTileRows);
    const float* t = cur; cur = nxt; nxt = (float*)t;   // ping-pong, no copy
  }

  const int gtotal = 3 * B * (DIM / 4);
  gather_out<<<(gtotal + blk - 1) / blk, blk, 0, stream>>>(acc, users, pos, neg, out, B);
}